// PacketOnlineModel_88287347737193
// MI455X (gfx1250) — compile-verified
//
#include <hip/hip_runtime.h>

typedef __attribute__((ext_vector_type(2))) float v2f;
typedef __attribute__((ext_vector_type(8))) float v8f;

#define B_TOT 2048
#define S_TOT 4096
#define DIM   8
#define VOCAB 19
#define CHUNK 128             // 2^7 steps per chunk
#define NC    (S_TOT / CHUNK) // 32 chunks

// lane l <-> lane l^16  (ds_swizzle group-of-32: and=0x1f, or=0, xor=0x10)
__device__ __forceinline__ float swz16(float x) {
    return __int_as_float(__builtin_amdgcn_ds_swizzle(__float_as_int(x), 0x401F));
}

// ---------------------------------------------------------------------------
// Phase 0a: embU[v][e] = sum_d embed[v,d]*w_in[e,d];  ru[b][e] = gate_b * sum_d res[b,d]*w_in[e,d]
// ---------------------------------------------------------------------------
__global__ __launch_bounds__(256)
void prep_kernel(const float* __restrict__ res, const float* __restrict__ gate,
                 const float* __restrict__ embed, const float* __restrict__ w_in,
                 float* __restrict__ embu, float* __restrict__ ru) {
    int g = blockIdx.x * blockDim.x + threadIdx.x;
    if (g < B_TOT * DIM) {
        int b = g >> 3, e = g & 7;
        float acc = 0.f;
        #pragma unroll
        for (int d = 0; d < 8; ++d) acc += res[b * 8 + d] * w_in[e * 8 + d];
        ru[g] = acc * gate[b];
    }
    if (g < VOCAB * DIM) {
        int v = g >> 3, e = g & 7;
        float acc = 0.f;
        #pragma unroll
        for (int d = 0; d < 8; ++d) acc += embed[v * 8 + d] * w_in[e * 8 + d];
        embu[g] = acc;
    }
}

// ---------------------------------------------------------------------------
// Phase 0b: M = w_state^CHUNK via 7 squarings of the 8x8 in LDS
// ---------------------------------------------------------------------------
__global__ __launch_bounds__(64)
void matpow_kernel(const float* __restrict__ w_state, float* __restrict__ Mout) {
    __shared__ float P[64];
    int t = threadIdx.x;           // 64 threads, one per (i,j)
    int i = t >> 3, j = t & 7;
    P[t] = w_state[t];
    __syncthreads();
    #pragma unroll
    for (int k = 0; k < 7; ++k) {  // 2^7 == CHUNK
        float acc = 0.f;
        #pragma unroll
        for (int d = 0; d < 8; ++d) acc += P[i * 8 + d] * P[d * 8 + j];
        __syncthreads();
        P[t] = acc;
        __syncthreads();
    }
    Mout[t] = P[t];
}

// ---------------------------------------------------------------------------
// Phases 1 & 3: one wave owns 16 batch rows x one chunk of 128 steps.
// State kept as (e x b) in WMMA C/D layout; A = w_state held in A-layout regs.
// Pad rows (M=8..15, upper-lane slots of C/D) are allowed to hold garbage:
// A's pad rows are zero and no consumer ever reads D's pad rows.
//   WRITE_OUT=false: start from zero state, emit chunk carry S_c.
//   WRITE_OUT=true : start from entry E_c, stream all h_t to d_out.
// ---------------------------------------------------------------------------
template <bool WRITE_OUT>
__global__ __launch_bounds__(32)
void rnn_chunk(const int* __restrict__ ids,
               const float* __restrict__ w_state,
               const float* __restrict__ embu,
               const float* __restrict__ ru,
               const float* __restrict__ entry,   // [B][NC][8] (phase 3)
               float* __restrict__ carry,         // [B][NC][8] (phase 1)
               float* __restrict__ out)           // [B][S][8]  (phase 3)
{
    __shared__ int s_ids[16 * CHUNK];
    __shared__ __align__(16) float s_embu[VOCAB * DIM];

    const int  lane = threadIdx.x;   // 0..31, single wave
    const int  m    = lane & 15;
    const int  half = lane >> 4;
    const int  b0   = blockIdx.x * 16;
    const int  c    = blockIdx.y;
    const long t0   = (long)c * CHUNK;

    // stage embU into LDS
    for (int i = lane; i < VOCAB * DIM; i += 32) s_embu[i] = embu[i];
    // stage ids for 16 rows x CHUNK steps (coalesced int4)
    for (int i = lane; i < 16 * (CHUNK / 4); i += 32) {
        int row = i >> 5;            // CHUNK/4 == 32
        int j4  = i & 31;
        const int4* src = (const int4*)(ids + (long)(b0 + row) * S_TOT + t0);
        ((int4*)s_ids)[row * 32 + j4] = src[j4];
    }

    // A = w_state in 16x4 A-layout; rows M=8..15 zero (not needed for D's real rows:
    // A's pad rows feed only D's pad rows; the guard keeps w_state reads in bounds).
    v2f a_lo, a_hi;
    {
        int  k0  = 2 * half;         // VGPR0 holds K=0 (lanes0-15) / K=2 (lanes16-31)
        bool act = (m < 8);
        a_lo.x = act ? w_state[m * 8 + k0    ] : 0.f;
        a_lo.y = act ? w_state[m * 8 + k0 + 1] : 0.f;
        a_hi.x = act ? w_state[m * 8 + k0 + 4] : 0.f;
        a_hi.y = act ? w_state[m * 8 + k0 + 5] : 0.f;
    }

    // per-batch-row constant input drive (upper lanes: harmless duplicate)
    float4 ru0, ru1;
    {
        const float4* rp = (const float4*)(ru + (long)(b0 + m) * 8);
        ru0 = rp[0]; ru1 = rp[1];
    }

    // initial state in D layout (upper-lane copies are unused pad garbage)
    v8f st;
    if (WRITE_OUT) {
        const float4* ep = (const float4*)(entry + ((long)(b0 + m) * NC + c) * 8);
        float4 e0 = ep[0], e1 = ep[1];
        st[0] = e0.x;  st[1] = e0.y;  st[2] = e0.z;  st[3] = e0.w;
        st[4] = e1.x;  st[5] = e1.y;  st[6] = e1.z;  st[7] = e1.w;
    } else {
        #pragma unroll
        for (int i = 0; i < 8; ++i) st[i] = 0.f;
    }
    __syncthreads();

    for (int j = 0; j < CHUNK; ++j) {
        // C = u_t = embU[id] + ru ; all 32 lanes load (upper = duplicate, unused)
        int id = s_ids[m * CHUNK + j];
        const float4* ep = (const float4*)&s_embu[id * DIM];
        float4 e0 = ep[0], e1 = ep[1];
        v8f cm;
        cm[0] = e0.x + ru0.x;  cm[1] = e0.y + ru0.y;
        cm[2] = e0.z + ru0.z;  cm[3] = e0.w + ru0.w;
        cm[4] = e1.x + ru1.x;  cm[5] = e1.y + ru1.y;
        cm[6] = e1.z + ru1.z;  cm[7] = e1.w + ru1.w;

        // D-layout state -> B-layout operands.
        // Swizzles are UNCONDITIONAL (full EXEC); selects merge halves after.
        float s2 = swz16(st[2]);
        float s3 = swz16(st[3]);
        float s6 = swz16(st[6]);
        float s7 = swz16(st[7]);
        v2f b_lo, b_hi;
        b_lo.x = half ? s2 : st[0];   // rows {0,2}
        b_lo.y = half ? s3 : st[1];   // rows {1,3}
        b_hi.x = half ? s6 : st[4];   // rows {4,6}
        b_hi.y = half ? s7 : st[5];   // rows {5,7}

        // st_new = w_state * st + u_t  (K=8 as two K=4 WMMA f32 ops)
        v8f acc = __builtin_amdgcn_wmma_f32_16x16x4_f32(
            false, a_lo, false, b_lo, (short)0, cm, false, false);
        acc = __builtin_amdgcn_wmma_f32_16x16x4_f32(
            false, a_hi, false, b_hi, (short)0, acc, false, false);
        st = acc;

        if (WRITE_OUT) {
            if (lane < 16) {   // lane m holds the full 8-vector for batch row b0+m
                float4* op = (float4*)(out + (((long)(b0 + m) * S_TOT) + t0 + j) * 8);
                op[0] = make_float4(st[0], st[1], st[2], st[3]);
                op[1] = make_float4(st[4], st[5], st[6], st[7]);
            }
        }
    }

    if (!WRITE_OUT) {
        if (lane < 16) {
            float4* cp = (float4*)(carry + ((long)(b0 + m) * NC + c) * 8);
            cp[0] = make_float4(st[0], st[1], st[2], st[3]);
            cp[1] = make_float4(st[4], st[5], st[6], st[7]);
        }
    }
}

// ---------------------------------------------------------------------------
// Phase 2: per batch row, scan chunk carries: E_0 = 0; E_{c+1} = M*E_c + S_c
// ---------------------------------------------------------------------------
__global__ __launch_bounds__(256)
void chunkscan_kernel(const float* __restrict__ M, const float* __restrict__ carry,
                      float* __restrict__ entry) {
    __shared__ float sM[64];
    if (threadIdx.x < 64) sM[threadIdx.x] = M[threadIdx.x];
    __syncthreads();
    int b = blockIdx.x * blockDim.x + threadIdx.x;
    if (b >= B_TOT) return;
    float e[8];
    #pragma unroll
    for (int i = 0; i < 8; ++i) e[i] = 0.f;
    for (int c = 0; c < NC; ++c) {
        float*       ep = entry + ((long)b * NC + c) * 8;
        const float* sp = carry + ((long)b * NC + c) * 8;
        float t[8];
        #pragma unroll
        for (int i = 0; i < 8; ++i) {
            ep[i] = e[i];
            float acc = sp[i];
            #pragma unroll
            for (int d = 0; d < 8; ++d) acc += sM[i * 8 + d] * e[d];
            t[i] = acc;
        }
        #pragma unroll
        for (int i = 0; i < 8; ++i) e[i] = t[i];
    }
}

// ---------------------------------------------------------------------------
extern "C" void kernel_launch(void* const* d_in, const int* in_sizes, int n_in,
                              void* d_out, int out_size, void* d_ws, size_t ws_size,
                              hipStream_t stream) {
    (void)in_sizes; (void)n_in; (void)out_size; (void)ws_size;
    const int*   ids     = (const int*)  d_in[0];
    const float* res     = (const float*)d_in[1];
    const float* gate    = (const float*)d_in[2];
    const float* embed   = (const float*)d_in[3];
    const float* w_in    = (const float*)d_in[4];
    const float* w_state = (const float*)d_in[5];
    float* out = (float*)d_out;
    float* ws  = (float*)d_ws;

    // workspace layout (floats), all 16B aligned:
    float* ws_embu  = ws;              // 152 used (pad 160)
    float* ws_M     = ws + 160;        // 64
    float* ws_ru    = ws + 224;        // B*8 = 16384
    float* ws_carry = ws + 16608;      // B*NC*8 = 524288
    float* ws_entry = ws + 540896;     // B*NC*8 = 524288   (total ~4.3 MB)

    prep_kernel<<<dim3(64), dim3(256), 0, stream>>>(res, gate, embed, w_in, ws_embu, ws_ru);
    matpow_kernel<<<dim3(1), dim3(64), 0, stream>>>(w_state, ws_M);
    rnn_chunk<false><<<dim3(B_TOT / 16, NC), dim3(32), 0, stream>>>(
        ids, w_state, ws_embu, ws_ru, nullptr, ws_carry, nullptr);
    chunkscan_kernel<<<dim3(B_TOT / 256), dim3(256), 0, stream>>>(ws_M, ws_carry, ws_entry);
    rnn_chunk<true><<<dim3(B_TOT / 16, NC), dim3(32), 0, stream>>>(
        ids, w_state, ws_embu, ws_ru, ws_entry, nullptr, out);
}